// RingGNNNet_76390288327378
// MI455X (gfx1250) — compile-verified
//
#include <hip/hip_runtime.h>
#include <hip/hip_bf16.h>

typedef __attribute__((ext_vector_type(2))) float v2f;
typedef __attribute__((ext_vector_type(8))) float v8f;

static constexpr int M     = 400;
static constexpr int P     = M * M;          // 160000 points (unpadded)
static constexpr int MS    = 416;            // padded matrix edge: 26*16 = 13*32
static constexpr int PS    = MS * MS;        // 173056 = 676*256
static constexpr int H     = 32;             // hidden / d_out
static constexpr int NE    = 40000;          // edges
static constexpr float LN_EPS = 1e-5f;

// ---------------------------------------------------------------------------
// Per-channel reductions: rows, cols, diag, sum_diag, total. One block per d.
// x is unpadded [D,400,400].
// ---------------------------------------------------------------------------
__global__ void reduce_kernel(const float* __restrict__ x,
                              float* __restrict__ diag, float* __restrict__ rows,
                              float* __restrict__ cols, float* __restrict__ sdiag,
                              float* __restrict__ tot) {
    const int d   = blockIdx.x;
    const int tid = threadIdx.x;
    const float* xd = x + (size_t)d * P;
    __shared__ float s1[256], s2[256];
    float locRow = 0.f, locDiag = 0.f;
    for (int i = tid; i < M; i += 256) {
        float s = 0.f;
        const float* r = xd + (size_t)i * M;
        for (int j = 0; j < M; ++j) s += r[j];
        rows[d * M + i] = s;
        locRow += s;
        float dv = xd[(size_t)i * M + i];
        diag[d * M + i] = dv;
        locDiag += dv;
    }
    for (int j = tid; j < M; j += 256) {
        float s = 0.f;
        for (int i = 0; i < M; ++i) s += xd[(size_t)i * M + j];
        cols[d * M + j] = s;
    }
    s1[tid] = locRow; s2[tid] = locDiag;
    __syncthreads();
    for (int off = 128; off > 0; off >>= 1) {
        if (tid < off) { s1[tid] += s1[tid + off]; s2[tid] += s2[tid + off]; }
        __syncthreads();
    }
    if (tid == 0) { tot[d] = s1[0]; sdiag[d] = s2[0]; }
}

// ---------------------------------------------------------------------------
// Low-rank pieces of the 15-basis contraction:
//   U[k,s,i], V[k,s,j], G[k,s,i] (diag-only), W[k,s] (constant)
// ---------------------------------------------------------------------------
__global__ void uvwg_kernel(const float* __restrict__ c0, const float* __restrict__ c1,
                            const float* __restrict__ c2,
                            const float* __restrict__ diag, const float* __restrict__ rows,
                            const float* __restrict__ cols, const float* __restrict__ sdiag,
                            const float* __restrict__ tot, int D,
                            float* __restrict__ U, float* __restrict__ V,
                            float* __restrict__ G, float* __restrict__ Wc) {
    int idx = blockIdx.x * 256 + threadIdx.x;
    if (idx >= 3 * H * M) return;
    int k = idx / (H * M);
    int r = idx - k * H * M;
    int s = r / M;
    int i = r - s * M;
    const float* C = (k == 0) ? c0 : ((k == 1) ? c1 : c2);
    const float invm  = 1.0f / (float)M;
    const float invm2 = invm * invm;
    float u = 0.f, v = 0.f, g = 0.f, w = 0.f;
    for (int d = 0; d < D; ++d) {
        const float* cb = C + ((size_t)d * H + s) * 15;
        float dv = diag[d * M + i], rv = rows[d * M + i], cv = cols[d * M + i];
        float sd = sdiag[d], tt = tot[d];
        u += cb[5]  * cv * invm + cb[6] * rv * invm + cb[11] * dv;
        v += cb[7]  * cv * invm + cb[8] * rv * invm + cb[12] * dv;
        g += cb[0]  * dv + cb[1] * sd * invm + cb[2] * rv * invm
           + cb[3]  * cv * invm + cb[4] * tt * invm2;
        w += cb[13] * sd * invm + cb[14] * tt * invm2;
    }
    U[(k * H + s) * M + i] = u;
    V[(k * H + s) * M + i] = v;
    G[(k * H + s) * M + i] = g;
    if (i == 0) Wc[k * H + s] = w;
}

// ---------------------------------------------------------------------------
// oi_k[s, i*416+j] for k=0..2 over the PADDED 416x416 domain.
// Pad region (i>=400 || j>=400) is written as zero so the GEMM K-padding is
// exact. Dense part: per-point cA*x + cB*x^T plus low-rank pieces + diag bias.
// ---------------------------------------------------------------------------
template <int D>
__global__ void oi_kernel(const float* __restrict__ x,
                          const float* __restrict__ c0, const float* __restrict__ c1,
                          const float* __restrict__ c2,
                          const float* __restrict__ db0, const float* __restrict__ db1,
                          const float* __restrict__ U, const float* __restrict__ V,
                          const float* __restrict__ G, const float* __restrict__ Wc,
                          float* __restrict__ oi) {
    __shared__ float cA[3 * D * H];
    __shared__ float cB[3 * D * H];
    const int tid = threadIdx.x;
    for (int t = tid; t < 3 * D * H; t += 256) {
        int k = t / (D * H);
        int r = t - k * D * H;
        int d = r >> 5;
        int s = r & 31;
        const float* C = (k == 0) ? c0 : ((k == 1) ? c1 : c2);
        size_t base = ((size_t)d * H + s) * 15;
        cA[t] = C[base + 9];   // op10: x
        cB[t] = C[base + 10];  // op11: x^T
    }
    __syncthreads();

    const int p = blockIdx.x * 256 + tid;   // grid = 676*256 == PS exactly
    const int i = p / MS;
    const int j = p - i * MS;

    if (i >= M || j >= M) {                 // K/M/N padding: must be zero
        for (int ks = 0; ks < 3 * H; ++ks) oi[(size_t)ks * PS + p] = 0.f;
        return;
    }

    float xv[D], xt[D];
#pragma unroll
    for (int d = 0; d < D; ++d) {
        xv[d] = x[(size_t)d * P + i * M + j];
        xt[d] = x[(size_t)d * P + j * M + i];
    }
    for (int k = 0; k < 3; ++k) {
        for (int s = 0; s < H; ++s) {
            float acc = U[(k * H + s) * M + i] + V[(k * H + s) * M + j] + Wc[k * H + s];
            if (i == j) {
                acc += G[(k * H + s) * M + i];
                if (k == 0) acc += db0[s];
                else if (k == 1) acc += db1[s];
            }
#pragma unroll
            for (int d = 0; d < D; ++d)
                acc += cA[(k * D + d) * H + s] * xv[d] + cB[(k * D + d) * H + s] * xt[d];
            oi[((size_t)k * H + s) * PS + p] = acc;
        }
    }
}

// ---------------------------------------------------------------------------
// Batched fp32 GEMM via V_WMMA_F32_16X16X4_F32, padded 416x416x416:
//   mm[s] = oi2[s] @ oi1b[s],  s = 0..31
// Register-blocked: one 32x32 output tile per wave (4 accumulators), so each
// A fragment feeds 2 WMMAs and each B fragment feeds 2 WMMAs -> 8 FLOP/byte
// from L2 (2x the naive tile). 13x13 wave tiles per channel, 676 blocks of
// 8 waves. K-pad rows/cols are zeros, so the 104-step K loop is uniform and
// EXEC stays all-ones around the WMMAs.
// ---------------------------------------------------------------------------
__global__ void gemm_wmma(const float* __restrict__ A_, const float* __restrict__ B_,
                          float* __restrict__ C_) {
    const int wave = threadIdx.x >> 5;
    const int lane = threadIdx.x & 31;
    const int t    = blockIdx.x * 8 + wave;      // < 32 * 169
    const int s    = t / 169;
    const int tile = t - s * 169;
    const int tm   = tile / 13, tn = tile - tm * 13;
    const int i0 = tm * 32, j0 = tn * 32;

    const float* A = A_ + (size_t)s * PS;
    const float* B = B_ + (size_t)s * PS;
    const int r  = lane & 15;
    const int kb = (lane >> 4) << 1;             // 0 or 2
    const float* arow0 = A + (size_t)(i0 + r) * MS;
    const float* arow1 = A + (size_t)(i0 + 16 + r) * MS;

    v8f acc00 = {}, acc01 = {}, acc10 = {}, acc11 = {};
    for (int k0 = 0; k0 < MS; k0 += 4) {
        const int kk = k0 + kb;
        const float* brow0 = B + (size_t)kk * MS;
        const float* brow1 = B + (size_t)(kk + 1) * MS;
        v2f a0, a1, b0, b1;
        a0[0] = arow0[kk];      a0[1] = arow0[kk + 1];
        a1[0] = arow1[kk];      a1[1] = arow1[kk + 1];
        b0[0] = brow0[j0 + r];      b0[1] = brow1[j0 + r];
        b1[0] = brow0[j0 + 16 + r]; b1[1] = brow1[j0 + 16 + r];
        // L2 prefetch ~16 K-steps ahead (global_prefetch_b8)
        __builtin_prefetch(arow0 + kk + 64, 0, 1);
        __builtin_prefetch(B + (size_t)(kk + 64) * MS + j0 + r, 0, 1);
        acc00 = __builtin_amdgcn_wmma_f32_16x16x4_f32(false, a0, false, b0, (short)0, acc00, false, false);
        acc01 = __builtin_amdgcn_wmma_f32_16x16x4_f32(false, a0, false, b1, (short)0, acc01, false, false);
        acc10 = __builtin_amdgcn_wmma_f32_16x16x4_f32(false, a1, false, b0, (short)0, acc10, false, false);
        acc11 = __builtin_amdgcn_wmma_f32_16x16x4_f32(false, a1, false, b1, (short)0, acc11, false, false);
    }
    const int rb = (lane >> 4) * 8;
    float* Cout = C_ + (size_t)s * PS;
#pragma unroll
    for (int v = 0; v < 8; ++v) {
        Cout[(size_t)(i0 + rb + v) * MS + j0 + r]           = acc00[v];
        Cout[(size_t)(i0 + rb + v) * MS + j0 + 16 + r]      = acc01[v];
        Cout[(size_t)(i0 + 16 + rb + v) * MS + j0 + r]      = acc10[v];
        Cout[(size_t)(i0 + 16 + rb + v) * MS + j0 + 16 + r] = acc11[v];
    }
}

// ---------------------------------------------------------------------------
// switch-combine + all_bias + LayerNorm(channel) + relu + residual linear.
// oi0/mm are padded (stride 416); xin/xout unpadded (stride 400).
// ---------------------------------------------------------------------------
template <int D>
__global__ void combine_kernel(const float* __restrict__ oi0, const float* __restrict__ mm,
                               const float* __restrict__ xin,
                               const float* __restrict__ psw0, const float* __restrict__ psw1,
                               const float* __restrict__ all_bias,
                               const float* __restrict__ ln_g, const float* __restrict__ ln_b,
                               const float* __restrict__ res_W, const float* __restrict__ res_b,
                               float* __restrict__ xout) {
    __shared__ float rw[D * H];
    __shared__ float sg[H], sb[H], sab[H], srb[H];
    const int tid = threadIdx.x;
    for (int t = tid; t < D * H; t += 256) rw[t] = res_W[t];   // res_W[e][d], e-major
    if (tid < H) { sg[tid] = ln_g[tid]; sb[tid] = ln_b[tid];
                   sab[tid] = all_bias[tid]; srb[tid] = res_b[tid]; }
    __syncthreads();
    const float sw0 = psw0[0], sw1 = psw1[0];

    const int p = blockIdx.x * 256 + tid;        // grid covers P exactly
    const int i = p / M;
    const int j = p - i * M;
    const size_t po = (size_t)i * MS + j;        // padded offset

    float pre[H];
    float mu = 0.f;
#pragma unroll
    for (int s = 0; s < H; ++s) {
        float v = sw0 * oi0[(size_t)s * PS + po] + sw1 * mm[(size_t)s * PS + po] + sab[s];
        pre[s] = v;
        mu += v;
    }
    mu *= (1.0f / H);
    float var = 0.f;
#pragma unroll
    for (int s = 0; s < H; ++s) { float dlt = pre[s] - mu; var += dlt * dlt; }
    var *= (1.0f / H);
    const float inv = rsqrtf(var + LN_EPS);

    float xv[D];
#pragma unroll
    for (int d = 0; d < D; ++d) xv[d] = xin[(size_t)d * P + p];

#pragma unroll
    for (int s = 0; s < H; ++s) {
        float y = (pre[s] - mu) * inv * sg[s] + sb[s];
        y = fmaxf(y, 0.f);
        float r = srb[s];
#pragma unroll
        for (int d = 0; d < D; ++d) r += rw[s * D + d] * xv[d];
        xout[(size_t)s * P + p] = y + r;
    }
}

// ---------------------------------------------------------------------------
// feats: column sums per stage -> node_feats[j][off+c]   (x unpadded)
// ---------------------------------------------------------------------------
__global__ void colsum_kernel(const float* __restrict__ x, int D, int off,
                              float* __restrict__ nf) {
    int t = blockIdx.x * 256 + threadIdx.x;
    if (t >= D * M) return;
    int c = t / M;
    int j = t - c * M;
    float s = 0.f;
    const float* xc = x + (size_t)c * P;
    for (int i = 0; i < M; ++i) s += xc[(size_t)i * M + j];
    nf[j * 68 + off + c] = s;
}

// ---------------------------------------------------------------------------
// Edge MLP: gather + 136->68 relu, 68->34 relu, 34->2
// ---------------------------------------------------------------------------
__global__ void mlp1_kernel(const float* __restrict__ nf, const int* __restrict__ el,
                            const float* __restrict__ W, const float* __restrict__ b,
                            float* __restrict__ h1) {
    int idx = blockIdx.x * 256 + threadIdx.x;
    if (idx >= NE * 68) return;
    int e = idx / 68;
    int o = idx - e * 68;
    const int srcn = el[e];
    const int dstn = el[NE + e];
    const float* ns = nf + srcn * 68;
    const float* nd = nf + dstn * 68;
    const float* w  = W + o * 136;
    float acc = b[o];
    for (int d = 0; d < 68; ++d) acc += w[d] * ns[d];
    for (int d = 0; d < 68; ++d) acc += w[68 + d] * nd[d];
    h1[(size_t)e * 68 + o] = fmaxf(acc, 0.f);
}

__global__ void mlp2_kernel(const float* __restrict__ h1, const float* __restrict__ W,
                            const float* __restrict__ b, float* __restrict__ h2) {
    int idx = blockIdx.x * 256 + threadIdx.x;
    if (idx >= NE * 34) return;
    int e = idx / 34;
    int o = idx - e * 34;
    const float* w = W + o * 68;
    const float* hi = h1 + (size_t)e * 68;
    float acc = b[o];
    for (int d = 0; d < 68; ++d) acc += w[d] * hi[d];
    h2[(size_t)e * 34 + o] = fmaxf(acc, 0.f);
}

__global__ void mlp3_kernel(const float* __restrict__ h2, const float* __restrict__ W,
                            const float* __restrict__ b, float* __restrict__ out) {
    int idx = blockIdx.x * 256 + threadIdx.x;
    if (idx >= NE * 2) return;
    int e = idx / 2;
    int c = idx - e * 2;
    const float* w = W + c * 34;
    const float* hi = h2 + (size_t)e * 34;
    float acc = b[c];
    for (int d = 0; d < 34; ++d) acc += w[d] * hi[d];
    out[(size_t)e * 2 + c] = acc;
}

// ---------------------------------------------------------------------------
// Host driver
// ---------------------------------------------------------------------------
extern "C" void kernel_launch(void* const* d_in, const int* in_sizes, int n_in,
                              void* d_out, int out_size, void* d_ws, size_t ws_size,
                              hipStream_t stream) {
    auto F = [&](int i) { return (const float*)d_in[i]; };
    const float* X0 = F(1);                  // x_with_edge_feat [4,400,400]
    const int*   EL = (const int*)d_in[2];   // edge_list [2, 40000]
    float* out = (float*)d_out;

    float* ws = (float*)d_ws;
    float* x1    = ws;                       // 32*P   (unpadded)
    float* x2    = x1 + (size_t)32 * P;      // 32*P   (unpadded)
    float* oi    = x2 + (size_t)32 * P;      // 96*PS  (padded, k-major)
    float* mm    = oi + (size_t)96 * PS;     // 32*PS  (padded)
    float* red   = mm + (size_t)32 * PS;
    float* diag  = red;                      // 32*400
    float* rows  = red + 32 * M;             // 32*400
    float* cols  = red + 64 * M;             // 32*400
    float* sdiag = red + 96 * M;             // 32
    float* tot   = sdiag + 32;               // 32
    float* uvg   = tot + 32;
    float* U = uvg, *V = U + 3 * H * M, *G = V + 3 * H * M, *Wc = G + 3 * H * M;
    // post-layer reuse of the oi region
    float* nf = oi;                          // 400*68
    float* h1 = oi + 32768;                  // 40000*68
    float* h2 = h1 + (size_t)NE * 68;        // 40000*34

    // ---------------- layer 0 (D = 4) ----------------
    {
        const float *c0 = F(3), *c1 = F(4), *c2 = F(5);
        const float *db0 = F(6), *db1 = F(7);
        const float *ab = F(9), *sw0 = F(10), *sw1 = F(11);
        const float *lg = F(12), *lb = F(13), *rW = F(14), *rb = F(15);
        reduce_kernel<<<4, 256, 0, stream>>>(X0, diag, rows, cols, sdiag, tot);
        uvwg_kernel<<<(3 * H * M + 255) / 256, 256, 0, stream>>>(
            c0, c1, c2, diag, rows, cols, sdiag, tot, 4, U, V, G, Wc);
        oi_kernel<4><<<PS / 256, 256, 0, stream>>>(X0, c0, c1, c2, db0, db1, U, V, G, Wc, oi);
        gemm_wmma<<<676, 256, 0, stream>>>(oi + (size_t)64 * PS, oi + (size_t)32 * PS, mm);
        combine_kernel<4><<<P / 256, 256, 0, stream>>>(oi, mm, X0, sw0, sw1, ab, lg, lb, rW, rb, x1);
    }
    // ---------------- layer 1 (D = 32) ----------------
    {
        const float *c0 = F(16), *c1 = F(17), *c2 = F(18);
        const float *db0 = F(19), *db1 = F(20);
        const float *ab = F(22), *sw0 = F(23), *sw1 = F(24);
        const float *lg = F(25), *lb = F(26), *rW = F(27), *rb = F(28);
        reduce_kernel<<<32, 256, 0, stream>>>(x1, diag, rows, cols, sdiag, tot);
        uvwg_kernel<<<(3 * H * M + 255) / 256, 256, 0, stream>>>(
            c0, c1, c2, diag, rows, cols, sdiag, tot, 32, U, V, G, Wc);
        oi_kernel<32><<<PS / 256, 256, 0, stream>>>(x1, c0, c1, c2, db0, db1, U, V, G, Wc, oi);
        gemm_wmma<<<676, 256, 0, stream>>>(oi + (size_t)64 * PS, oi + (size_t)32 * PS, mm);
        combine_kernel<32><<<P / 256, 256, 0, stream>>>(oi, mm, x1, sw0, sw1, ab, lg, lb, rW, rb, x2);
    }
    // ---------------- features + edge MLP ----------------
    colsum_kernel<<<(4 * M + 255) / 256, 256, 0, stream>>>(X0, 4, 0, nf);
    colsum_kernel<<<(32 * M + 255) / 256, 256, 0, stream>>>(x1, 32, 4, nf);
    colsum_kernel<<<(32 * M + 255) / 256, 256, 0, stream>>>(x2, 32, 36, nf);
    mlp1_kernel<<<(NE * 68 + 255) / 256, 256, 0, stream>>>(nf, EL, F(29), F(30), h1);
    mlp2_kernel<<<(NE * 34 + 255) / 256, 256, 0, stream>>>(h1, F(31), F(32), h2);
    mlp3_kernel<<<(NE * 2 + 255) / 256, 256, 0, stream>>>(h2, F(33), F(34), out);
}